// MAB_67568425501167
// MI455X (gfx1250) — compile-verified
//
#include <hip/hip_runtime.h>

typedef __attribute__((ext_vector_type(2))) float v2f;
typedef __attribute__((ext_vector_type(8))) float v8f;

#define T_DIM 40
#define C_DIM 64
#define INTER 16
#define NSUB 3
#define NPAD 48      // t-dimension padded to 3 WMMA N-tiles
#define HEADS 8
#define DS 320
#define NPT 25
#define SQ 10
#define BATCH 256

// ---------------------------------------------------------------------------
// Kernel 1: per-(b,v) GEMM  a/b = W(16x64) @ X(64x40) + bias, for 3 subsets.
// x in (B,V,64,40) layout; aout/bout in (B,V,3,16,40).
// Jobs: 6 M-tiles (3 subsets x {a,b}) x 3 N-tiles = 18, looped per wave.
// ---------------------------------------------------------------------------
__global__ void gcn_ab_kernel(const float* __restrict__ x,
                              const float* __restrict__ Wa,  // (3,16,64)
                              const float* __restrict__ ba,  // (3,16)
                              const float* __restrict__ Wb,
                              const float* __restrict__ bbp,
                              float* __restrict__ aout,
                              float* __restrict__ bout,
                              int V) {
  __shared__ float Xs[C_DIM * NPAD];
  const int bv  = blockIdx.x;
  const int tid = threadIdx.x;
  const float* xs = x + (size_t)bv * (C_DIM * T_DIM);
  for (int j = tid; j < C_DIM * NPAD; j += blockDim.x) {
    int c = j / NPAD, t = j - c * NPAD;
    Xs[j] = (t < T_DIM) ? xs[c * T_DIM + t] : 0.0f;
  }
  __syncthreads();

  const int lane   = tid & 31;
  const int wave   = tid >> 5;
  const int nwaves = blockDim.x >> 5;
  const int r      = lane & 15;
  const int hi     = lane >> 4;
  const int koff   = hi * 2;

  for (int job = wave; job < 18; job += nwaves) {
    int mt  = job / 3;
    int nt  = job - mt * 3;
    int s   = mt >> 1;
    int isB = mt & 1;
    const float* W    = (isB ? Wb : Wa) + s * (INTER * C_DIM);
    const float* bias = (isB ? bbp : ba) + s * INTER;
    int n0 = nt * 16;
    v8f acc = {};
    for (int k = 0; k < C_DIM; k += 4) {
      v2f af, bf;
      af.x = W[r * C_DIM + k + koff];
      af.y = W[r * C_DIM + k + koff + 1];
      bf.x = Xs[(k + koff) * NPAD + n0 + r];
      bf.y = Xs[(k + koff + 1) * NPAD + n0 + r];
      acc = __builtin_amdgcn_wmma_f32_16x16x4_f32(false, af, false, bf,
                                                  (short)0, acc, false, false);
    }
    float* out = (isB ? bout : aout) + ((size_t)bv * NSUB + s) * (INTER * T_DIM);
    int t = n0 + r;
    if (t < T_DIM) {
      for (int rr = 0; rr < 8; ++rr) {
        int i = rr + 8 * hi;
        out[i * T_DIM + t] = acc[rr] + bias[i];
      }
    }
  }
}

// ---------------------------------------------------------------------------
// Kernel 2: S(b,s,v1,v2) = softmax_v1( dot640(a[b,v1,s], b[b,v2,s]) / 640 ) + PA
// One block per (b, subset).
// ---------------------------------------------------------------------------
__global__ void gcn_s_kernel(const float* __restrict__ abuf,  // (B,V,3,640)
                             const float* __restrict__ bbuf,
                             const float* __restrict__ PA,    // (3,V,V)
                             float* __restrict__ S,           // (B,3,V,V)
                             int V) {
  __shared__ float Sm[NPT * NPT];
  const int b   = blockIdx.x / NSUB;
  const int s   = blockIdx.x - b * NSUB;
  const int tid = threadIdx.x;
  const float invden = 1.0f / (float)(INTER * T_DIM);
  for (int j = tid; j < V * V; j += blockDim.x) {
    int v1 = j / V, v2 = j - v1 * V;
    const float* ap = abuf + (((size_t)b * V + v1) * NSUB + s) * (INTER * T_DIM);
    const float* bp = bbuf + (((size_t)b * V + v2) * NSUB + s) * (INTER * T_DIM);
    float acc = 0.0f;
    for (int k = 0; k < INTER * T_DIM; k += 4) {
      float4 a4 = *reinterpret_cast<const float4*>(ap + k);
      float4 b4 = *reinterpret_cast<const float4*>(bp + k);
      acc += a4.x * b4.x + a4.y * b4.y + a4.z * b4.z + a4.w * b4.w;
    }
    Sm[j] = acc * invden;
  }
  __syncthreads();
  if (tid < V) {
    int v2 = tid;
    float m = -3.4e38f;
    for (int v1 = 0; v1 < V; ++v1) m = fmaxf(m, Sm[v1 * V + v2]);
    float sum = 0.0f;
    for (int v1 = 0; v1 < V; ++v1) sum += __expf(Sm[v1 * V + v2] - m);
    float inv = 1.0f / sum;
    float* sp = S + ((size_t)b * NSUB + s) * V * V;
    const float* pap = PA + (size_t)s * V * V;
    for (int v1 = 0; v1 < V; ++v1)
      sp[v1 * V + v2] = __expf(Sm[v1 * V + v2] - m) * inv + pap[v1 * V + v2];
  }
}

// ---------------------------------------------------------------------------
// Kernel 3: fused vertex-mix (x @ S) + Wd GEMM + bias + BN + residual + ReLU.
// One block per (b, v_out). z_s staged in LDS, then 64x64 @ 64x40 WMMA x3 sub.
// ---------------------------------------------------------------------------
__global__ void gcn_zd_kernel(const float* __restrict__ x,   // (B,V,2560)
                              const float* __restrict__ S,   // (B,3,V,V)
                              const float* __restrict__ Wd,  // (3,64,64)
                              const float* __restrict__ bd,  // (3,64)
                              const float* __restrict__ gamma,
                              const float* __restrict__ beta,
                              float* __restrict__ y,         // (B,V,2560)
                              int V) {
  __shared__ float Zs[NSUB * C_DIM * NPAD];   // 36 KB
  __shared__ float Scol[NSUB * NPT];
  __shared__ float bsum[C_DIM], scale[C_DIM], shift[C_DIM];
  const int bv   = blockIdx.x;
  const int b    = bv / V;
  const int vout = bv - b * V;
  const int tid  = threadIdx.x;

  if (tid < NSUB * V) {
    int s = tid / V, u = tid - s * V;
    Scol[s * NPT + u] = S[(((size_t)b * NSUB + s) * V + u) * V + vout];
  }
  if (tid < C_DIM) {
    bsum[tid]  = bd[tid] + bd[C_DIM + tid] + bd[2 * C_DIM + tid];
    scale[tid] = gamma[tid] * rsqrtf(1.0f + 1e-5f);
    shift[tid] = beta[tid];
  }
  for (int j = tid; j < NSUB * C_DIM * NPAD; j += blockDim.x) Zs[j] = 0.0f;
  __syncthreads();

  const float* xb = x + (size_t)b * V * (C_DIM * T_DIM);
  for (int e = tid; e < C_DIM * T_DIM; e += blockDim.x) {
    int c = e / T_DIM, t = e - c * T_DIM;
    float a0 = 0.f, a1 = 0.f, a2 = 0.f;
    for (int u = 0; u < V; ++u) {
      float xv = xb[(size_t)u * (C_DIM * T_DIM) + e];
      a0 += Scol[u] * xv;
      a1 += Scol[NPT + u] * xv;
      a2 += Scol[2 * NPT + u] * xv;
    }
    int zi = c * NPAD + t;
    Zs[zi]                    = a0;
    Zs[C_DIM * NPAD + zi]     = a1;
    Zs[2 * C_DIM * NPAD + zi] = a2;
  }
  __syncthreads();

  const int lane   = tid & 31;
  const int wave   = tid >> 5;
  const int nwaves = blockDim.x >> 5;
  const int r      = lane & 15;
  const int hi     = lane >> 4;
  const int koff   = hi * 2;

  for (int job = wave; job < 12; job += nwaves) {   // 4 M-tiles x 3 N-tiles
    int mt = job / 3, nt = job - mt * 3;
    int m0 = mt * 16, n0 = nt * 16;
    v8f acc = {};
    for (int s = 0; s < NSUB; ++s) {
      const float* W = Wd + s * (C_DIM * C_DIM);
      const float* Z = Zs + s * (C_DIM * NPAD);
      for (int k = 0; k < C_DIM; k += 4) {
        v2f af, bf;
        af.x = W[(m0 + r) * C_DIM + k + koff];
        af.y = W[(m0 + r) * C_DIM + k + koff + 1];
        bf.x = Z[(k + koff) * NPAD + n0 + r];
        bf.y = Z[(k + koff + 1) * NPAD + n0 + r];
        acc = __builtin_amdgcn_wmma_f32_16x16x4_f32(false, af, false, bf,
                                                    (short)0, acc, false, false);
      }
    }
    int t = n0 + r;
    if (t < T_DIM) {
      float* yp = y + (size_t)bv * (C_DIM * T_DIM);
      const float* xp = x + (size_t)bv * (C_DIM * T_DIM);
      for (int rr = 0; rr < 8; ++rr) {
        int o = m0 + rr + 8 * hi;
        float val = (acc[rr] + bsum[o]) * scale[o] + shift[o] + xp[o * T_DIM + t];
        yp[o * T_DIM + t] = fmaxf(val, 0.0f);
      }
    }
  }
}

// ---------------------------------------------------------------------------
// Kernel 4: 8-head attention. One block per (b, q).
// ---------------------------------------------------------------------------
__global__ void attn_kernel(const float* __restrict__ Q,   // (B,10,2560)
                            const float* __restrict__ Kf,  // (B,25,2560)
                            const float* __restrict__ Vf,
                            float* __restrict__ O) {
  __shared__ float sc[HEADS * NPT];
  const int bq  = blockIdx.x;
  const int b   = bq / SQ;
  const int tid = threadIdx.x;
  const float* qp = Q + (size_t)bq * (HEADS * DS);
  const float invs = rsqrtf((float)(HEADS * DS));
  if (tid < HEADS * NPT) {
    int h = tid / NPT, k = tid - h * NPT;
    const float* kp = Kf + ((size_t)b * NPT + k) * (HEADS * DS) + h * DS;
    const float* qh = qp + h * DS;
    float acc = 0.0f;
    for (int d = 0; d < DS; d += 4) {
      float4 q4 = *(const float4*)(qh + d);
      float4 k4 = *(const float4*)(kp + d);
      acc += q4.x * k4.x + q4.y * k4.y + q4.z * k4.z + q4.w * k4.w;
    }
    sc[tid] = acc * invs;
  }
  __syncthreads();
  if (tid < HEADS) {
    float m = -3.4e38f;
    for (int k = 0; k < NPT; ++k) m = fmaxf(m, sc[tid * NPT + k]);
    float sum = 0.0f;
    for (int k = 0; k < NPT; ++k) {
      float e = __expf(sc[tid * NPT + k] - m);
      sc[tid * NPT + k] = e;
      sum += e;
    }
    float inv = 1.0f / sum;
    for (int k = 0; k < NPT; ++k) sc[tid * NPT + k] *= inv;
  }
  __syncthreads();
  float* op = O + (size_t)bq * (HEADS * DS);
  for (int e = tid; e < HEADS * DS; e += blockDim.x) {
    int h = e / DS;
    float acc = qp[e];
    const float* at = sc + h * NPT;
    for (int k = 0; k < NPT; ++k)
      acc += at[k] * Vf[((size_t)b * NPT + k) * (HEADS * DS) + e];
    op[e] = acc;
  }
}

// ---------------------------------------------------------------------------
// Kernel 5: out = O + relu(Og)
// ---------------------------------------------------------------------------
__global__ void final_kernel(const float* __restrict__ O,
                             const float* __restrict__ Og,
                             float* __restrict__ out, int n) {
  int i = blockIdx.x * blockDim.x + threadIdx.x;
  if (i < n) out[i] = O[i] + fmaxf(Og[i], 0.0f);
}

// ---------------------------------------------------------------------------
extern "C" void kernel_launch(void* const* d_in, const int* in_sizes, int n_in,
                              void* d_out, int out_size, void* d_ws, size_t ws_size,
                              hipStream_t stream) {
  const float* Q = (const float*)d_in[0];
  const float* K = (const float*)d_in[1];
  // Each group: PA, Wa, ba, Wb, bb, Wd, bd, gamma, beta
  const float* fck[9]; const float* fcv[9]; const float* fco[9];
  for (int i = 0; i < 9; ++i) {
    fck[i] = (const float*)d_in[2 + i];
    fcv[i] = (const float*)d_in[11 + i];
    fco[i] = (const float*)d_in[20 + i];
  }
  const int B = BATCH, VK = NPT, VO = SQ;
  float* ws = (float*)d_ws;
  size_t off = 0;
  float* Kg  = ws + off; off += (size_t)B * VK * 2560;
  float* Vg  = ws + off; off += (size_t)B * VK * 2560;
  float* Obf = ws + off; off += (size_t)B * VO * 2560;
  float* Og  = ws + off; off += (size_t)B * VO * 2560;
  float* abf = ws + off; off += (size_t)B * VK * NSUB * (INTER * T_DIM);
  float* bbf = ws + off; off += (size_t)B * VK * NSUB * (INTER * T_DIM);
  float* Sbf = ws + off; off += (size_t)B * NSUB * VK * VK;

  auto run_gcn = [&](const float* x, const float* const* P, float* yout, int V) {
    gcn_ab_kernel<<<B * V, 128, 0, stream>>>(x, P[1], P[2], P[3], P[4], abf, bbf, V);
    gcn_s_kernel<<<B * NSUB, 256, 0, stream>>>(abf, bbf, P[0], Sbf, V);
    gcn_zd_kernel<<<B * V, 256, 0, stream>>>(x, Sbf, P[5], P[6], P[7], P[8], yout, V);
  };

  run_gcn(K, fck, Kg, VK);                       // Kg = unit_gcn(Kr, fck)
  run_gcn(K, fcv, Vg, VK);                       // Vg = unit_gcn(Kr, fcv)
  attn_kernel<<<B * VO, 256, 0, stream>>>(Q, Kg, Vg, Obf);
  run_gcn(Obf, fco, Og, VO);                     // Og = unit_gcn(O, fco)
  int n = B * VO * 2560;
  final_kernel<<<(n + 255) / 256, 256, 0, stream>>>(Obf, Og, (float*)d_out, n);
}